// FCCaps_68685116998276
// MI455X (gfx1250) — compile-verified
//
#include <hip/hip_runtime.h>
#include <hip/hip_bf16.h>
#include <math.h>

// Problem constants (from reference)
#define B_   128
#define I_   2048
#define J_   6      // OUT_CAPS
#define O_   16     // Cout
#define C_   32     // Cin
#define NJO  96     // J_*O_
#define MAX_ITERS 16
#define XPAD 36     // LDS row stride (floats): 144B rows -> 16B aligned, no bank conflicts

typedef float v2f __attribute__((ext_vector_type(2)));
typedef float v8f __attribute__((ext_vector_type(8)));

struct RState { float loss; int done; };

// ---------------------------------------------------------------------------
// Kernel 1: u_hat[b,i,j,o] = sum_c W1[0,i,j,o,c] * x[b,i,c]
// Per block: one i. GEMM  (128x32) x (32x96) -> (128x96) via f32 WMMA 16x16x4.
// ---------------------------------------------------------------------------
__global__ __launch_bounds__(256) void uhat_kernel(const float* __restrict__ x,
                                                   const float* __restrict__ W1,
                                                   float* __restrict__ u_hat) {
    __shared__ float Xs[128 * XPAD];
    __shared__ float Wsm[96 * XPAD];
    const int i   = blockIdx.x;
    const int tid = threadIdx.x;

    // Stage X_i (128x32) as float4: 1024 vec4 elements, 4 per thread
#pragma unroll
    for (int it = 0; it < 4; ++it) {
        const int idx = tid + it * 256;          // 0..1023
        const int b = idx >> 3, c4 = idx & 7;
        const float4 val = *(const float4*)&x[(b * I_ + i) * C_ + c4 * 4];
        *(float4*)&Xs[b * XPAD + c4 * 4] = val;
    }
    // Stage W_i (96x32): W1 is (1, I, 8, 16, 32); first 3072 floats per i-block
    // are exactly j < OUT_CAPS. 768 vec4 elements, 3 per thread.
#pragma unroll
    for (int it = 0; it < 3; ++it) {
        const int idx = tid + it * 256;          // 0..767
        const int n = idx >> 3, c4 = idx & 7;
        const float4 val = *(const float4*)&W1[i * (8 * 16 * 32) + n * 32 + c4 * 4];
        *(float4*)&Wsm[n * XPAD + c4 * 4] = val;
    }
    __syncthreads();

    const int wave   = tid >> 5;
    const int lane   = tid & 31;
    const int laneLo = lane & 15;
    const int laneHi = lane >> 4;

    // 48 output tiles (8 M-tiles x 6 N-tiles), 8 waves -> 6 tiles each.
    for (int q = 0; q < 6; ++q) {
        const int T  = wave + 8 * q;     // 0..47
        const int mt = T & 7;            // M tile (b)
        const int nt = T >> 3;           // N tile (== j, N-tile width 16)
        const float* Xrow = &Xs[(mt * 16 + laneLo) * XPAD];
        const float* Wrow = &Wsm[(nt * 16 + laneLo) * XPAD];

        // Batch ALL operand loads up front so ds loads pipeline ahead of WMMAs
        float ax[16], bx[16];
#pragma unroll
        for (int kk = 0; kk < 8; ++kk) {
            const int k0 = kk * 4 + 2 * laneHi;
            ax[2 * kk]     = Xrow[k0];
            ax[2 * kk + 1] = Xrow[k0 + 1];
            bx[2 * kk]     = Wrow[k0];
            bx[2 * kk + 1] = Wrow[k0 + 1];
        }
        // Two independent accumulator chains to break WMMA RAW dependency
        v8f acc0 = {}, acc1 = {};
#pragma unroll
        for (int kk = 0; kk < 8; kk += 2) {
            v2f a0; a0[0] = ax[2 * kk];     a0[1] = ax[2 * kk + 1];
            v2f b0; b0[0] = bx[2 * kk];     b0[1] = bx[2 * kk + 1];
            v2f a1; a1[0] = ax[2 * kk + 2]; a1[1] = ax[2 * kk + 3];
            v2f b1; b1[0] = bx[2 * kk + 2]; b1[1] = bx[2 * kk + 3];
            acc0 = __builtin_amdgcn_wmma_f32_16x16x4_f32(
                false, a0, false, b0, (short)0, acc0, false, false);
            acc1 = __builtin_amdgcn_wmma_f32_16x16x4_f32(
                false, a1, false, b1, (short)0, acc1, false, false);
        }
        const v8f acc = acc0 + acc1;

        // C/D layout: VGPR r holds (m = r + 8*laneHi, n = laneLo)
        const size_t base =
            ((size_t)(mt * 16 + 8 * laneHi) * I_ + i) * NJO + nt * 16 + laneLo;
        const size_t bstride = (size_t)I_ * NJO;   // stride between b rows
#pragma unroll
        for (int r = 0; r < 8; ++r) {
            u_hat[base + (size_t)r * bstride] = acc[r];
        }
    }
}

// ---------------------------------------------------------------------------
// Init: b_ij = 0, routing state = {loss=0, done=0}
// ---------------------------------------------------------------------------
__global__ void init_kernel(float* __restrict__ bij, RState* st) {
    const int t = blockIdx.x * blockDim.x + threadIdx.x;
    if (t < B_ * I_ * J_) bij[t] = 0.0f;
    if (t == 0) { st->loss = 0.0f; st->done = 0; }
}

// ---------------------------------------------------------------------------
// R1: per (b, i-block of 128): softmax over j -> c_un; partial reductions
//     csum_part[b,blk,j] = sum_i c_un ; Su_part[b,blk,j,o] = sum_i c_un*u_hat
// ---------------------------------------------------------------------------
__global__ __launch_bounds__(128) void route_c_kernel(
        const float* __restrict__ u_hat, const float* __restrict__ bij,
        float* __restrict__ c_un, float* __restrict__ csum_part,
        float* __restrict__ Su_part, const RState* __restrict__ st) {
    if (st->done) return;                     // uniform across grid
    __shared__ float c_sh[128 * 6];
    const int b   = blockIdx.x >> 4;
    const int blk = blockIdx.x & 15;
    const int i0  = blk * 128;
    const int tid = threadIdx.x;              // 0..127

    {   // softmax over j for i = i0 + tid
        const int i = i0 + tid;
        const float* brow = &bij[(b * I_ + i) * J_];
        float m = brow[0];
#pragma unroll
        for (int j = 1; j < J_; ++j) m = fmaxf(m, brow[j]);
        float cl[J_]; float s = 0.0f;
#pragma unroll
        for (int j = 0; j < J_; ++j) { cl[j] = __expf(brow[j] - m); s += cl[j]; }
        const float inv = 1.0f / s;
#pragma unroll
        for (int j = 0; j < J_; ++j) {
            cl[j] *= inv;
            c_sh[tid * J_ + j] = cl[j];
            c_un[(b * I_ + i) * J_ + j] = cl[j];
        }
    }
    __syncthreads();

    if (tid < J_) {   // csum partial for this i-block
        float s = 0.0f;
        for (int il = 0; il < 128; ++il) s += c_sh[il * J_ + tid];
        csum_part[(b * 16 + blk) * J_ + tid] = s;
    }
    if (tid < NJO) {  // Su partial: jo = tid
        const int j = tid >> 4;
        const float* up = &u_hat[(size_t)(b * I_ + i0) * NJO + tid];
        float acc = 0.0f;
        for (int il = 0; il < 128; ++il) {
            if (il + 16 < 128)
                __builtin_prefetch(up + (size_t)(il + 16) * NJO, 0, 1);
            acc += c_sh[il * J_ + j] * up[(size_t)il * NJO];
        }
        Su_part[(b * 16 + blk) * NJO + tid] = acc;
    }
}

// ---------------------------------------------------------------------------
// R2: finish reductions, v[b,j,:] = squash(Su/csum); store csum[b,j]
// ---------------------------------------------------------------------------
__global__ __launch_bounds__(256) void route_v_kernel(
        const float* __restrict__ csum_part, const float* __restrict__ Su_part,
        float* __restrict__ csum, float* __restrict__ v,
        const RState* __restrict__ st) {
    if (st->done) return;
    const int t = blockIdx.x * 256 + threadIdx.x;   // 0..12287 -> (b, j, o)
    const int b = t / NJO, jo = t % NJO, j = jo >> 4, o = jo & 15;
    float cs = 0.0f, su = 0.0f;
    for (int blk = 0; blk < 16; ++blk) {
        cs += csum_part[(b * 16 + blk) * J_ + j];
        su += Su_part[(b * 16 + blk) * NJO + jo];
    }
    const float S = su / cs;
    __shared__ float sm[256];
    sm[threadIdx.x] = S * S;
    __syncthreads();
    const int g = threadIdx.x & ~15;     // 16 consecutive threads share (b,j)
    float s2 = 0.0f;
#pragma unroll
    for (int k = 0; k < 16; ++k) s2 += sm[g + k];
    const float scale = sqrtf(s2) / (0.5f + s2);
    v[t] = scale * S;
    if (o == 0) csum[b * J_ + j] = cs;
}

// ---------------------------------------------------------------------------
// R3: per (b,i,j): dd = 1 - ||u_sq - v||^2 (via s2, u.v, ||v||^2);
//     bij += dd; block-partial of sum(c*dd)
// ---------------------------------------------------------------------------
__global__ __launch_bounds__(256) void route_dd_kernel(
        const float* __restrict__ u_hat, const float* __restrict__ c_un,
        const float* __restrict__ csum, const float* __restrict__ v,
        float* __restrict__ bij, float* __restrict__ kde_part,
        const RState* __restrict__ st) {
    if (st->done) return;
    const int t = blockIdx.x * 256 + threadIdx.x;   // 0..1572863 -> (b,i,j)
    const int j = t % J_;
    const int bi = t / J_;
    const int b = bi >> 11;                         // /2048
    const float* u  = &u_hat[(size_t)t * O_];       // ((b*I+i)*J + j)*16 == t*16
    const float* vp = &v[(b * J_ + j) * O_];
    float s2 = 0.0f, uv = 0.0f, vv = 0.0f;
#pragma unroll
    for (int o = 0; o < O_; ++o) {
        const float uo = u[o], vo = vp[o];
        s2 += uo * uo; uv += uo * vo; vv += vo * vo;
    }
    const float s  = sqrtf(s2) / (0.5f + s2);
    const float dd = 1.0f - (s * s * s2 - 2.0f * s * uv + vv);
    bij[t] += dd;
    const float cval = c_un[t] / csum[b * J_ + j];
    __shared__ float sm[256];
    sm[threadIdx.x] = cval * dd;
    __syncthreads();
    for (int stp = 128; stp > 0; stp >>= 1) {
        if (threadIdx.x < stp) sm[threadIdx.x] += sm[threadIdx.x + stp];
        __syncthreads();
    }
    if (threadIdx.x == 0) kde_part[blockIdx.x] = sm[0];
}

// ---------------------------------------------------------------------------
// R4: kde = log(sum/B); convergence check; update state
// ---------------------------------------------------------------------------
__global__ __launch_bounds__(256) void route_kde_kernel(
        const float* __restrict__ kde_part, RState* st, int nparts) {
    if (st->done) return;
    float s = 0.0f;
    for (int k = threadIdx.x; k < nparts; k += 256) s += kde_part[k];
    __shared__ float sm[256];
    sm[threadIdx.x] = s;
    __syncthreads();
    for (int stp = 128; stp > 0; stp >>= 1) {
        if (threadIdx.x < stp) sm[threadIdx.x] += sm[threadIdx.x + stp];
        __syncthreads();
    }
    if (threadIdx.x == 0) {
        const float kde = logf(sm[0] / (float)B_);
        const int conv = fabsf(kde - st->loss) < 0.05f;
        st->loss = kde;
        if (conv) st->done = 1;
    }
}

// ---------------------------------------------------------------------------
// Outputs: poses (B,6,16,1) then activations (B,6,1)
// ---------------------------------------------------------------------------
__global__ __launch_bounds__(256) void out_kernel(const float* __restrict__ v,
                                                  float* __restrict__ out) {
    const int t = blockIdx.x * 256 + threadIdx.x;
    if (t < B_ * J_ * O_) out[t] = v[t];
    if (t < B_ * J_) {
        const float* vp = &v[t * O_];
        float s = 0.0f;
#pragma unroll
        for (int o = 0; o < O_; ++o) s += vp[o] * vp[o];
        out[B_ * J_ * O_ + t] = sqrtf(s);
    }
}

// ---------------------------------------------------------------------------
extern "C" void kernel_launch(void* const* d_in, const int* in_sizes, int n_in,
                              void* d_out, int out_size, void* d_ws, size_t ws_size,
                              hipStream_t stream) {
    const float* x  = (const float*)d_in[0];   // (128, 2048, 32)
    // d_in[1] = y (int, unused by the reference computation)
    const float* W1 = (const float*)d_in[2];   // (1, 2048, 8, 16, 32)
    float* out = (float*)d_out;

    // Workspace layout (floats); total ~28.54M floats (~114 MB)
    float* ws        = (float*)d_ws;
    float* u_hat     = ws;                                     // 25,165,824
    float* bij       = u_hat + (size_t)B_ * I_ * J_ * O_;      //  1,572,864
    float* c_un      = bij + (size_t)B_ * I_ * J_;             //  1,572,864
    float* csum_part = c_un + (size_t)B_ * I_ * J_;            //     12,288
    float* Su_part   = csum_part + (size_t)B_ * 16 * J_;       //    196,608
    float* csum      = Su_part + (size_t)B_ * 16 * NJO;        //        768
    float* v         = csum + (size_t)B_ * J_;                 //     12,288
    float* kde_part  = v + (size_t)B_ * J_ * O_;               //      6,144
    RState* st       = (RState*)(kde_part + 6144);

    // 1) u_hat via WMMA
    uhat_kernel<<<I_, 256, 0, stream>>>(x, W1, u_hat);

    // 2) routing init
    init_kernel<<<(B_ * I_ * J_ + 255) / 256, 256, 0, stream>>>(bij, st);

    // 3) 16 routing iterations (kernels self-gate on st->done)
    const int nddblocks = (B_ * I_ * J_) / 256;   // 6144
    for (int it = 0; it < MAX_ITERS; ++it) {
        route_c_kernel<<<B_ * 16, 128, 0, stream>>>(u_hat, bij, c_un,
                                                    csum_part, Su_part, st);
        route_v_kernel<<<(B_ * NJO) / 256, 256, 0, stream>>>(csum_part, Su_part,
                                                             csum, v, st);
        route_dd_kernel<<<nddblocks, 256, 0, stream>>>(u_hat, c_un, csum, v,
                                                       bij, kde_part, st);
        route_kde_kernel<<<1, 256, 0, stream>>>(kde_part, st, nddblocks);
    }

    // 4) outputs
    out_kernel<<<48, 256, 0, stream>>>(v, out);
}